// GATLevelPredictor_26104811225560
// MI455X (gfx1250) — compile-verified
//
#include <hip/hip_runtime.h>
#include <hip/hip_bf16.h>

// ---------------------------------------------------------------------------
// GAT level predictor for MI455X (gfx1250, wave32).
//  - dense GEMMs via V_WMMA_F32_16X16X4_F32 (full f32 precision)
//  - edge softmax via ordered-uint atomic max + f32 atomic adds
// ---------------------------------------------------------------------------

typedef __attribute__((ext_vector_type(2))) float v2f;
typedef __attribute__((ext_vector_type(8))) float v8f;

#define HC 128   // heads * hidden = 2*64
#define CH 64    // hidden channels
#define NH 2     // heads

// ---- monotone float <-> uint mapping for atomic max on floats --------------
__device__ __forceinline__ unsigned f2ord(float f) {
    unsigned u = __float_as_uint(f);
    return (u & 0x80000000u) ? ~u : (u | 0x80000000u);
}
__device__ __forceinline__ float ord2f(unsigned o) {
    unsigned u = (o & 0x80000000u) ? (o & 0x7fffffffu) : ~o;
    return __uint_as_float(u);
}
__device__ __forceinline__ float leaky02(float v) { return v >= 0.f ? v : 0.2f * v; }

// ---------------------------------------------------------------------------
// GEMM: Out[n,128] = A[n,128] @ W[128,128]   (row-major, f32 WMMA)
// grid.x = ceil(n/16) M-tiles, 256 threads = 8 waves = 8 column tiles.
// ---------------------------------------------------------------------------
__global__ __launch_bounds__(256) void gat_gemm_wmma(
    const float* __restrict__ A, const float* __restrict__ W,
    float* __restrict__ Out, int n)
{
    const int wave = threadIdx.x >> 5;      // column tile 0..7
    const int lane = threadIdx.x & 31;
    const int half = lane >> 4;             // 0: K lo pair, 1: K hi pair
    const int l16  = lane & 15;
    const int tileM = blockIdx.x;

    int row = tileM * 16 + l16;
    int rowc = row < n ? row : (n - 1);     // clamp loads; EXEC stays all-1s
    const int col = wave * 16 + l16;

    const float* __restrict__ arow = A + (size_t)rowc * HC;
    __builtin_prefetch(arow, 0, 0);         // global_prefetch_b8

    v8f acc = {};
#pragma unroll
    for (int k = 0; k < HC; k += 4) {
        const int ka = k + 2 * half;        // this lane's K pair
        // A 16x4 fragment: lanes 0-15 -> K=0,1 ; lanes 16-31 -> K=2,3
        v2f a = *(const v2f*)(arow + ka);
        // B 4x16 fragment (mirrored): v0 = W[ka][col], v1 = W[ka+1][col]
        v2f b;
        b.x = W[(size_t)ka * HC + col];
        b.y = W[(size_t)(ka + 1) * HC + col];
        acc = __builtin_amdgcn_wmma_f32_16x16x4_f32(
            /*neg_a=*/false, a, /*neg_b=*/false, b,
            /*c_mod=*/(short)0, acc, /*reuse_a=*/false, /*reuse_b=*/false);
    }

    // C/D layout: VGPR v holds M = half*8 + v for this lane's column.
    const int rbase = tileM * 16 + half * 8;
#pragma unroll
    for (int v = 0; v < 8; ++v) {
        int r = rbase + v;
        if (r < n) Out[(size_t)r * HC + col] = acc[v];
    }
}

// ---------------------------------------------------------------------------
// Attention coefficients: als[n,h] = sum_c xs[n,h,c]*a_src[h,c]; same for ald.
// One wave per node; lanes 0-15 handle head 0, lanes 16-31 head 1.
// ---------------------------------------------------------------------------
__global__ __launch_bounds__(256) void gat_attn_coef(
    const float* __restrict__ xs, const float* __restrict__ a_s,
    const float* __restrict__ a_d,
    float* __restrict__ als, float* __restrict__ ald, int n)
{
    const int node = blockIdx.x * 8 + (threadIdx.x >> 5);
    const int lane = threadIdx.x & 31;
    if (node >= n) return;

    const float4 x4 = ((const float4*)(xs + (size_t)node * HC))[lane];
    const float4 s4 = ((const float4*)a_s)[lane];
    const float4 d4 = ((const float4*)a_d)[lane];

    float ps = x4.x * s4.x + x4.y * s4.y + x4.z * s4.z + x4.w * s4.w;
    float pd = x4.x * d4.x + x4.y * d4.y + x4.z * d4.z + x4.w * d4.w;
#pragma unroll
    for (int o = 1; o < 16; o <<= 1) {      // reduce within each 16-lane half
        ps += __shfl_xor(ps, o, 32);
        pd += __shfl_xor(pd, o, 32);
    }
    if ((lane & 15) == 0) {
        const int h = lane >> 4;
        als[node * NH + h] = ps;
        ald[node * NH + h] = pd;
    }
}

// ---------------------------------------------------------------------------
// Edge pass 1: segment max of leaky(al_s[src]+al_d[dst]) into ordered uints.
// ---------------------------------------------------------------------------
__global__ __launch_bounds__(256) void gat_edge_max(
    const int* __restrict__ ei, int E, int ET,
    const float* __restrict__ als, const float* __restrict__ ald,
    unsigned* __restrict__ mord)
{
    const int e = blockIdx.x * 256 + threadIdx.x;
    if (e >= ET) return;
    int s, d;
    if (e < E) { s = ei[e]; d = ei[E + e]; } else { s = d = e - E; }
#pragma unroll
    for (int h = 0; h < NH; ++h) {
        float v = leaky02(als[s * NH + h] + ald[d * NH + h]);
        atomicMax(mord + d * NH + h, f2ord(v));
    }
}

// ---------------------------------------------------------------------------
// Edge pass 2: ex = exp(e - max[dst]); store per-edge, atomic-add denominator.
// ---------------------------------------------------------------------------
__global__ __launch_bounds__(256) void gat_edge_expsum(
    const int* __restrict__ ei, int E, int ET,
    const float* __restrict__ als, const float* __restrict__ ald,
    const unsigned* __restrict__ mord,
    float* __restrict__ ex, float* __restrict__ den)
{
    const int e = blockIdx.x * 256 + threadIdx.x;
    if (e >= ET) return;
    int s, d;
    if (e < E) { s = ei[e]; d = ei[E + e]; } else { s = d = e - E; }
#pragma unroll
    for (int h = 0; h < NH; ++h) {
        float v = leaky02(als[s * NH + h] + ald[d * NH + h]);
        float m = ord2f(mord[d * NH + h]);
        float x = __expf(v - m);
        ex[(size_t)e * NH + h] = x;
        atomicAdd(den + d * NH + h, x);
    }
}

// ---------------------------------------------------------------------------
// Edge pass 3: out[dst] += alpha * xs[src].  One wave per edge, float4/lane.
// ---------------------------------------------------------------------------
__global__ __launch_bounds__(256) void gat_edge_aggregate(
    const int* __restrict__ ei, int E, int ET,
    const float* __restrict__ xs, const float* __restrict__ ex,
    const float* __restrict__ den, float* __restrict__ out)
{
    const int e = blockIdx.x * 8 + (threadIdx.x >> 5);
    if (e >= ET) return;
    const int lane = threadIdx.x & 31;
    int s, d;
    if (e < E) { s = ei[e]; d = ei[E + e]; } else { s = d = e - E; }

    const int h = lane >> 4;                    // lanes 0-15: head0, 16-31: head1
    const float alpha = ex[(size_t)e * NH + h] / (den[d * NH + h] + 1e-16f);

    const float4 x4 = ((const float4*)(xs + (size_t)s * HC))[lane];
    float* o = out + (size_t)d * HC + lane * 4;
    atomicAdd(o + 0, alpha * x4.x);
    atomicAdd(o + 1, alpha * x4.y);
    atomicAdd(o + 2, alpha * x4.z);
    atomicAdd(o + 3, alpha * x4.w);
}

// ---------------------------------------------------------------------------
// Finalize layers 1/2 (concat=True): h = prelu(agg + b), in place.
// ---------------------------------------------------------------------------
__global__ __launch_bounds__(256) void gat_finalize_concat(
    float* __restrict__ agg, const float* __restrict__ b,
    const float* __restrict__ prelu_a, long total)
{
    long i = (long)blockIdx.x * 256 + threadIdx.x;
    if (i >= total) return;
    float v = agg[i] + b[i & (HC - 1)];
    float a = *prelu_a;
    agg[i] = v >= 0.f ? v : a * v;
}

// ---------------------------------------------------------------------------
// Layer 3 finalize (mean over heads) + level predictor, fused. Wave per node.
// ---------------------------------------------------------------------------
__global__ __launch_bounds__(256) void gat_mean_predict(
    const float* __restrict__ agg, const float* __restrict__ b3,
    const float* __restrict__ lpw, const float* __restrict__ lpb,
    const float* __restrict__ prelu_a, float* __restrict__ out, int n)
{
    const int node = blockIdx.x * 8 + (threadIdx.x >> 5);
    const int lane = threadIdx.x & 31;
    if (node >= n) return;
    const float a = *prelu_a;
    float accv = 0.f;
#pragma unroll
    for (int t = 0; t < 2; ++t) {
        const int c = lane + 32 * t;            // c in [0,64)
        float v = 0.5f * (agg[(size_t)node * HC + c] +
                          agg[(size_t)node * HC + CH + c]) + b3[c];
        v = v >= 0.f ? v : a * v;
        accv += v * lpw[c];
    }
#pragma unroll
    for (int o = 1; o < 32; o <<= 1) accv += __shfl_xor(accv, o, 32);
    if (lane == 0) out[node] = accv + lpb[0];
}

// ---------------------------------------------------------------------------
extern "C" void kernel_launch(void* const* d_in, const int* in_sizes, int n_in,
                              void* d_out, int out_size, void* d_ws, size_t ws_size,
                              hipStream_t stream)
{
    const float* x    = (const float*)d_in[0];
    const int*   ei   = (const int*)  d_in[1];
    const float* W1   = (const float*)d_in[2];
    const float* as1  = (const float*)d_in[3];
    const float* ad1  = (const float*)d_in[4];
    const float* b1   = (const float*)d_in[5];
    const float* W2   = (const float*)d_in[6];
    const float* as2  = (const float*)d_in[7];
    const float* ad2  = (const float*)d_in[8];
    const float* b2   = (const float*)d_in[9];
    const float* W3   = (const float*)d_in[10];
    const float* as3  = (const float*)d_in[11];
    const float* ad3  = (const float*)d_in[12];
    const float* b3   = (const float*)d_in[13];
    const float* lpw  = (const float*)d_in[14];
    const float* lpb  = (const float*)d_in[15];
    const float* pa   = (const float*)d_in[16];

    const int n  = in_sizes[0] / HC;       // 50000
    const int E  = in_sizes[1] / 2;        // 800000
    const int ET = E + n;                  // + self loops

    // ---- carve workspace ----
    char* p = (char*)d_ws;
    auto carve = [&](size_t bytes) -> void* {
        void* r = (void*)p;
        p += (bytes + 255) & ~(size_t)255;
        return r;
    };
    float*    xs   = (float*)   carve((size_t)n * HC * sizeof(float));
    float*    agg  = (float*)   carve((size_t)n * HC * sizeof(float));
    float*    als  = (float*)   carve((size_t)n * NH * sizeof(float));
    float*    ald  = (float*)   carve((size_t)n * NH * sizeof(float));
    unsigned* mord = (unsigned*)carve((size_t)n * NH * sizeof(unsigned));
    float*    den  = (float*)   carve((size_t)n * NH * sizeof(float));
    float*    ex   = (float*)   carve((size_t)ET * NH * sizeof(float));

    const int gTiles = (n + 15) / 16;
    const int gNode8 = (n + 7) / 8;
    const int gEdge  = (ET + 255) / 256;
    const int gEdge8 = (ET + 7) / 8;
    const long total = (long)n * HC;
    const int gElem  = (int)((total + 255) / 256);

    auto run_layer = [&](const float* in, const float* Wm,
                         const float* a_s, const float* a_d) {
        gat_gemm_wmma<<<gTiles, 256, 0, stream>>>(in, Wm, xs, n);
        gat_attn_coef<<<gNode8, 256, 0, stream>>>(xs, a_s, a_d, als, ald, n);
        hipMemsetAsync(mord, 0, (size_t)n * NH * sizeof(unsigned), stream); // ord 0 == -inf class
        hipMemsetAsync(den,  0, (size_t)n * NH * sizeof(float), stream);
        hipMemsetAsync(agg,  0, (size_t)n * HC * sizeof(float), stream);    // after gemm consumed it
        gat_edge_max      <<<gEdge,  256, 0, stream>>>(ei, E, ET, als, ald, mord);
        gat_edge_expsum   <<<gEdge,  256, 0, stream>>>(ei, E, ET, als, ald, mord, ex, den);
        gat_edge_aggregate<<<gEdge8, 256, 0, stream>>>(ei, E, ET, xs, ex, den, agg);
    };

    // layer 1
    run_layer(x, W1, as1, ad1);
    gat_finalize_concat<<<gElem, 256, 0, stream>>>(agg, b1, pa, total);
    // layer 2
    run_layer(agg, W2, as2, ad2);
    gat_finalize_concat<<<gElem, 256, 0, stream>>>(agg, b2, pa, total);
    // layer 3 (mean over heads) + predictor
    run_layer(agg, W3, as3, ad3);
    gat_mean_predict<<<gNode8, 256, 0, stream>>>(agg, b3, lpw, lpb, pa, (float*)d_out, n);
}